// LIFNode_25159918420498
// MI455X (gfx1250) — compile-verified
//
#include <hip/hip_runtime.h>
#include <cstdint>

// LIF forward: v = v*DECAY + x_t; s = (v >= V_TH); v -= s*V_TH  (soft reset)
// HBM-bound streaming kernel (512 MB @ 23.3 TB/s ~= 22 us floor).
// CDNA5 async global->LDS ring buffer for deep prefetch across the serial T loop;
// all timestep addressing folded into 24-bit instruction offsets.

typedef float v4f __attribute__((ext_vector_type(4)));

static constexpr int   T_STEPS   = 128;
static constexpr int   D_DIM     = 8192;
static constexpr int   ROW_BYTES = D_DIM * 4;      // 32768 B between timesteps
static constexpr float DECAY     = 0.5f;           // 1 - 1/tau, tau = 2
static constexpr float V_TH      = 1.0f;

static constexpr int BLOCK = 256;                  // 8 wave32s
static constexpr int PIPE  = 4;                    // LDS ring depth (prefetch distance 3)
static constexpr int BUF_BYTES = BLOCK * 16;       // 4096 B per ring slot

template <int OFF_BYTES>
__device__ __forceinline__ void async_load_b128(uint32_t lds_addr, const float* gaddr) {
    // vdst = per-lane LDS byte address, vaddr = 64-bit global base (GV mode),
    // timestep selected via immediate offset; NT: stream, don't thrash L2.
    asm volatile("global_load_async_to_lds_b128 %0, %1, off offset:%2 th:TH_LOAD_NT"
                 :: "v"(lds_addr), "v"(gaddr), "n"(OFF_BYTES)
                 : "memory");
}

template <int N>
__device__ __forceinline__ void wait_asynccnt() {
    asm volatile("s_wait_asynccnt %0" :: "n"(N) : "memory");
}

__global__ __launch_bounds__(BLOCK)
void lif_forward_kernel(const float* __restrict__ x,
                        float* __restrict__ out,
                        long long n_cols4) {
    __shared__ v4f smem[PIPE][BLOCK];              // 16 KB ring buffer

    const int tid = threadIdx.x;
    const long long col4 = (long long)blockIdx.x * BLOCK + tid;
    if (col4 >= n_cols4) return;

    const long long col = col4 * 4;                // flat (n, d) float index
    const long long n   = col / D_DIM;
    const long long d   = col - n * D_DIM;
    const float* xc = x   + n * (long long)T_STEPS * D_DIM + d;
    float*       oc = out + n * (long long)T_STEPS * D_DIM + d;

    // LDS byte address of this lane's slot in buffer 0
    const uint32_t lds_tid = (uint32_t)(uintptr_t)(&smem[0][tid]);

    v4f v = {0.0f, 0.0f, 0.0f, 0.0f};

    auto step = [&](int buf, float* optr) {
        v4f xv = smem[buf][tid];                   // ds_load_b128 (const LDS offset)
        v = v * DECAY + xv;
        v4f s;
        s.x = (v.x >= V_TH) ? 1.0f : 0.0f;
        s.y = (v.y >= V_TH) ? 1.0f : 0.0f;
        s.z = (v.z >= V_TH) ? 1.0f : 0.0f;
        s.w = (v.w >= V_TH) ? 1.0f : 0.0f;
        v = v - s * V_TH;                          // soft reset (forward only)
        __builtin_nontemporal_store(s, (v4f*)optr);
    };

    // Prologue: prefetch t = 0,1,2 into ring slots 0,1,2.
    async_load_b128<0 * ROW_BYTES>(lds_tid + 0 * BUF_BYTES, xc);
    async_load_b128<1 * ROW_BYTES>(lds_tid + 1 * BUF_BYTES, xc);
    async_load_b128<2 * ROW_BYTES>(lds_tid + 2 * BUF_BYTES, xc);

    // Main loop: t = 4*it + k. Issue t+3 (slot (k+3)&3, immediate offset),
    // wait until <=3 outstanding (in-order completion => t's slot landed), consume t.
#pragma unroll 1
    for (int it = 0; it < (T_STEPS / 4) - 1; ++it) {   // t = 0 .. 123
        async_load_b128<3 * ROW_BYTES>(lds_tid + 3 * BUF_BYTES, xc);
        wait_asynccnt<3>(); step(0, oc + 0 * D_DIM);
        async_load_b128<4 * ROW_BYTES>(lds_tid + 0 * BUF_BYTES, xc);
        wait_asynccnt<3>(); step(1, oc + 1 * D_DIM);
        async_load_b128<5 * ROW_BYTES>(lds_tid + 1 * BUF_BYTES, xc);
        wait_asynccnt<3>(); step(2, oc + 2 * D_DIM);
        async_load_b128<6 * ROW_BYTES>(lds_tid + 2 * BUF_BYTES, xc);
        wait_asynccnt<3>(); step(3, oc + 3 * D_DIM);
        xc += 4 * (long long)D_DIM;
        oc += 4 * (long long)D_DIM;
    }

    // Epilogue: xc/oc point at t = 124; loads 124..126 already issued.
    async_load_b128<3 * ROW_BYTES>(lds_tid + 3 * BUF_BYTES, xc);   // t = 127 -> slot 3
    wait_asynccnt<3>(); step(0, oc + 0 * D_DIM);                   // t = 124
    wait_asynccnt<2>(); step(1, oc + 1 * D_DIM);                   // t = 125
    wait_asynccnt<1>(); step(2, oc + 2 * D_DIM);                   // t = 126
    wait_asynccnt<0>(); step(3, oc + 3 * D_DIM);                   // t = 127
}

extern "C" void kernel_launch(void* const* d_in, const int* in_sizes, int n_in,
                              void* d_out, int out_size, void* d_ws, size_t ws_size,
                              hipStream_t stream) {
    (void)n_in; (void)out_size; (void)d_ws; (void)ws_size;
    const float* x   = (const float*)d_in[0];
    float*       out = (float*)d_out;

    // in_sizes[0] = N*T*D; float4 columns over (n, d): N*D/4
    const long long total   = (long long)in_sizes[0];
    const long long n_cols4 = total / ((long long)T_STEPS * 4);
    const int grid = (int)((n_cols4 + BLOCK - 1) / BLOCK);

    lif_forward_kernel<<<grid, BLOCK, 0, stream>>>(x, out, n_cols4);
}